// CNN_Baseline_62646392980178
// MI455X (gfx1250) — compile-verified
//
#include <hip/hip_runtime.h>

typedef __attribute__((ext_vector_type(2))) float v2f;
typedef __attribute__((ext_vector_type(8))) float v8f;

#define PVAL 16384
#define BATCH 8
#define FEAT_ROWS 72   // 67 feature rows padded to 72 (K multiple of 8), rows 67..71 zero
#define EPSV 1e-5f

#if defined(__AMDGCN__) && __has_builtin(__builtin_amdgcn_global_load_async_to_lds_b32)
#define GL_ASYNC 1
#else
#define GL_ASYNC 0
#endif

// ---------------------------------------------------------------------------
// Conv3d evaluated ONLY at gathered voxels (floor(pt_loc)), fused with feature
// assembly: feats[b, 0..2, p] = pt_loc, feats[b, 3..34, p] = conv+bias.
// ---------------------------------------------------------------------------
__global__ void __launch_bounds__(256) conv_gather_kernel(
    const float* __restrict__ x, const float* __restrict__ pt,
    const float* __restrict__ w3d, const float* __restrict__ b3d,
    float* __restrict__ feats)
{
  int gid = blockIdx.x * 256 + threadIdx.x;   // exactly B*P threads
  int b = gid >> 14;
  int p = gid & (PVAL - 1);
  float px = pt[gid * 3 + 0];
  float py = pt[gid * 3 + 1];
  float pz = pt[gid * 3 + 2];
  int ix = min(max((int)floorf(px), 0), 63);
  int iy = min(max((int)floorf(py), 0), 63);
  int iz = min(max((int)floorf(pz), 0), 63);

  float v[4][27];
#pragma unroll
  for (int ci = 0; ci < 4; ++ci) {
    const float* xb = x + (((size_t)b * 4 + ci) << 18);
#pragma unroll
    for (int dd = 0; dd < 3; ++dd)
#pragma unroll
      for (int hh = 0; hh < 3; ++hh)
#pragma unroll
        for (int ww = 0; ww < 3; ++ww) {
          int X = ix + dd - 1, Y = iy + hh - 1, Z = iz + ww - 1;
          bool ok = ((unsigned)X < 64u) && ((unsigned)Y < 64u) && ((unsigned)Z < 64u);
          v[ci][dd * 9 + hh * 3 + ww] = ok ? xb[((X << 6) + Y) * 64 + Z] : 0.f;
        }
  }

  float* fb = feats + (size_t)b * FEAT_ROWS * PVAL;
  fb[0 * PVAL + p] = px;
  fb[1 * PVAL + p] = py;
  fb[2 * PVAL + p] = pz;

  for (int co = 0; co < 32; ++co) {
    float a = b3d[co];
    const float* wr = w3d + co * 108;
#pragma unroll
    for (int ci = 0; ci < 4; ++ci)
#pragma unroll
      for (int t = 0; t < 27; ++t)
        a = fmaf(v[ci][t], wr[ci * 27 + t], a);
    fb[(size_t)(3 + co) * PVAL + p] = a;
  }
}

// ---------------------------------------------------------------------------
// Global max-pool over points: one block per (channel, batch).
// ---------------------------------------------------------------------------
__global__ void __launch_bounds__(256) glob_max_kernel(
    const float* __restrict__ feats, float* __restrict__ glob)
{
  int c = blockIdx.x;   // 0..31
  int b = blockIdx.y;   // 0..7
  const float* row = feats + ((size_t)b * FEAT_ROWS + 3 + c) * PVAL;
  float m = -3.402823466e38f;
  for (int p = threadIdx.x; p < PVAL; p += 256) m = fmaxf(m, row[p]);
  __shared__ float sm[256];
  sm[threadIdx.x] = m;
  __syncthreads();
  for (int s = 128; s > 0; s >>= 1) {
    if (threadIdx.x < s) sm[threadIdx.x] = fmaxf(sm[threadIdx.x], sm[threadIdx.x + s]);
    __syncthreads();
  }
  if (threadIdx.x == 0) glob[b * 32 + c] = sm[0];
}

// Broadcast glob into feats rows 35..66 and zero the K-pad rows 67..71.
__global__ void __launch_bounds__(256) glob_bcast_kernel(
    const float* __restrict__ glob, float* __restrict__ feats)
{
  int idx = blockIdx.x * 256 + threadIdx.x;   // B * 37 * P threads
  int p = idx & (PVAL - 1);
  int r = (idx >> 14) % 37;                   // 0..31 glob rows, 32..36 zero pad
  int b = idx / (37 * PVAL);
  float val = (r < 32) ? glob[b * 32 + r] : 0.f;
  feats[((size_t)b * FEAT_ROWS + 35 + r) * PVAL + p] = val;
}

// Zero-pad weights into Wp[Mpad x Kpad] so the GEMM loop is guard-free.
__global__ void __launch_bounds__(256) pack_w_kernel(
    const float* __restrict__ W, float* __restrict__ Wp,
    int O, int K, int Mpad, int Kpad)
{
  int idx = blockIdx.x * 256 + threadIdx.x;
  if (idx >= Mpad * Kpad) return;
  int m = idx / Kpad, k = idx % Kpad;
  Wp[idx] = (m < O && k < K) ? W[m * K + k] : 0.f;
}

// ---------------------------------------------------------------------------
// Y[b,o,n] = sum_k Wp[o,k] * X[b,k,n] + bias[o]  via V_WMMA_F32_16X16X4_F32.
// Block = 256 threads = 8 waves arranged MW(M) x NW(N); wave tile 16M x 64N.
// X tile (8K x NT) staged in LDS (double buffered, async DMA when available),
// stored COLUMN-MAJOR with stride 10 floats: each B fragment (k,k+1) pair is
// one aligned ds_load_b64 straight into the WMMA register pair; stride 10 is
// bank-conflict-free for both fragment reads and cooperative writes.
// ---------------------------------------------------------------------------
template <int MW>
__global__ void __launch_bounds__(256) gemm_lds_wmma(
    const float* __restrict__ Wp, const float* __restrict__ bias,
    const float* __restrict__ X, float* __restrict__ Y,
    int O, int Kpad, int Xalloc, int Yalloc)
{
  constexpr int NW = 8 / MW;
  constexpr int NT = NW * 64;       // block N tile
  constexpr int CS = 10;            // LDS column stride (floats), 8 k's + 2 pad
  constexpr int NE = NW * 2;        // tile elements per thread
  __shared__ __align__(16) float xs[2][NT * CS];

  const int tid   = threadIdx.x;
  const int lane  = tid & 31;
  const int wid   = tid >> 5;
  const int bz    = blockIdx.z;
  const int mw    = wid % MW;
  const int nw    = wid / MW;
  const int m0    = blockIdx.y * (MW * 16) + mw * 16;
  const int nbase = blockIdx.x * NT;
  const int lcol  = lane & 15;
  const int khalf = (lane < 16) ? 0 : 2;

  const float* Xb = X + (size_t)bz * Xalloc * PVAL + nbase;
  const float* Wr = Wp + (size_t)(m0 + lcol) * Kpad + khalf;

  v8f c0 = {}, c1 = {}, c2 = {}, c3 = {};

  // per-thread cooperative-load offsets (row-major global -> col-major LDS)
  size_t goff[NE];
  int    lidx[NE];
#pragma unroll
  for (int e = 0; e < NE; ++e) {
    int idx = tid + e * 256;
    int row = idx / NT, col = idx % NT;
    goff[e] = (size_t)row * PVAL + col;
    lidx[e] = col * CS + row;
  }

  auto compute = [&](int k0, const float* xtile) {
    v2f a0 = *(const v2f*)(Wr + k0);          // W rows k0+khalf, +khalf+1
    v2f a1 = *(const v2f*)(Wr + k0 + 4);
    const float* xbp = xtile + (nw * 64 + lcol) * CS + khalf;
    v2f b00 = *(const v2f*)(xbp + 0);
    v2f b10 = *(const v2f*)(xbp + 4);
    v2f b01 = *(const v2f*)(xbp + 16 * CS);
    v2f b11 = *(const v2f*)(xbp + 16 * CS + 4);
    v2f b02 = *(const v2f*)(xbp + 32 * CS);
    v2f b12 = *(const v2f*)(xbp + 32 * CS + 4);
    v2f b03 = *(const v2f*)(xbp + 48 * CS);
    v2f b13 = *(const v2f*)(xbp + 48 * CS + 4);
    c0 = __builtin_amdgcn_wmma_f32_16x16x4_f32(false, a0, false, b00, (short)0, c0, false, false);
    c1 = __builtin_amdgcn_wmma_f32_16x16x4_f32(false, a0, false, b01, (short)0, c1, false, false);
    c2 = __builtin_amdgcn_wmma_f32_16x16x4_f32(false, a0, false, b02, (short)0, c2, false, false);
    c3 = __builtin_amdgcn_wmma_f32_16x16x4_f32(false, a0, false, b03, (short)0, c3, false, false);
    c0 = __builtin_amdgcn_wmma_f32_16x16x4_f32(false, a1, false, b10, (short)0, c0, false, false);
    c1 = __builtin_amdgcn_wmma_f32_16x16x4_f32(false, a1, false, b11, (short)0, c1, false, false);
    c2 = __builtin_amdgcn_wmma_f32_16x16x4_f32(false, a1, false, b12, (short)0, c2, false, false);
    c3 = __builtin_amdgcn_wmma_f32_16x16x4_f32(false, a1, false, b13, (short)0, c3, false, false);
  };

#if GL_ASYNC
  auto stage_async = [&](const float* src, float* dst) {
#pragma unroll
    for (int e = 0; e < NE; ++e)
      __builtin_amdgcn_global_load_async_to_lds_b32(
          (int*)(src + goff[e]), (int*)(dst + lidx[e]), 0, 0);
  };
#endif

  // preload K-tile 0
#if GL_ASYNC
  stage_async(Xb, xs[0]);
  asm volatile("s_wait_asynccnt 0x0" ::: "memory");
#else
  {
    float pre[NE];
#pragma unroll
    for (int e = 0; e < NE; ++e) pre[e] = Xb[goff[e]];
#pragma unroll
    for (int e = 0; e < NE; ++e) xs[0][lidx[e]] = pre[e];
  }
#endif
  __syncthreads();

  int buf = 0;
  const float* Xpre = Xb + (size_t)8 * PVAL;
  for (int k0 = 0; k0 + 8 < Kpad; k0 += 8) {     // steady state, branch-free
#if GL_ASYNC
    stage_async(Xpre, xs[buf ^ 1]);
    compute(k0, xs[buf]);
    asm volatile("s_wait_asynccnt 0x0" ::: "memory");
#else
    float pre[NE];
#pragma unroll
    for (int e = 0; e < NE; ++e) pre[e] = Xpre[goff[e]];
    compute(k0, xs[buf]);
#pragma unroll
    for (int e = 0; e < NE; ++e) xs[buf ^ 1][lidx[e]] = pre[e];
#endif
    __syncthreads();
    buf ^= 1;
    Xpre += (size_t)8 * PVAL;
  }
  compute(Kpad - 8, xs[buf]);                    // peeled tail

  float* Yb = Y + (size_t)bz * Yalloc * PVAL + nbase + nw * 64 + lcol;
  const int mb = m0 + ((lane < 16) ? 0 : 8);
#pragma unroll
  for (int r = 0; r < 8; ++r) {
    int m = mb + r;
    if (m < O) {
      float bv = bias[m];
      size_t base = (size_t)m * PVAL;
      Yb[base +  0] = c0[r] + bv;
      Yb[base + 16] = c1[r] + bv;
      Yb[base + 32] = c2[r] + bv;
      Yb[base + 48] = c3[r] + bv;
    }
  }
}

// Per-channel training-mode BN statistics: one block per output channel.
__global__ void __launch_bounds__(256) stats_kernel(
    const float* __restrict__ Y, int alloc,
    float* __restrict__ mean, float* __restrict__ invstd)
{
  int o = blockIdx.x;
  float s1 = 0.f, s2 = 0.f;
  for (int b = 0; b < BATCH; ++b) {
    const float* row = Y + ((size_t)b * alloc + o) * PVAL;
    for (int p = threadIdx.x; p < PVAL; p += 256) {
      float v = row[p];
      s1 += v; s2 += v * v;
    }
  }
  __shared__ float sh1[256], sh2[256];
  sh1[threadIdx.x] = s1; sh2[threadIdx.x] = s2;
  __syncthreads();
  for (int s = 128; s > 0; s >>= 1) {
    if (threadIdx.x < s) {
      sh1[threadIdx.x] += sh1[threadIdx.x + s];
      sh2[threadIdx.x] += sh2[threadIdx.x + s];
    }
    __syncthreads();
  }
  if (threadIdx.x == 0) {
    float n   = (float)(BATCH * PVAL);
    float m   = sh1[0] / n;
    float var = sh2[0] / n - m * m;
    mean[o]   = m;
    invstd[o] = rsqrtf(fmaxf(var, 0.f) + EPSV);
  }
}

// In-place BN + ReLU.  O is a power of two for all uses here.
__global__ void __launch_bounds__(256) bnrelu_kernel(
    float* __restrict__ Y, const float* __restrict__ g, const float* __restrict__ beta,
    const float* __restrict__ mean, const float* __restrict__ invstd,
    int O, int alloc)
{
  size_t idx = (size_t)blockIdx.x * 256 + threadIdx.x;   // < B*O*P
  int p = (int)(idx & (PVAL - 1));
  int o = (int)((idx >> 14) % O);
  int b = (int)(idx / ((size_t)O * PVAL));
  size_t a = ((size_t)b * alloc + o) * PVAL + p;
  float v = Y[a];
  v = fmaf(g[o] * invstd[o], v - mean[o], beta[o]);
  Y[a] = fmaxf(v, 0.f);
}

// ---------------------------------------------------------------------------
extern "C" void kernel_launch(void* const* d_in, const int* in_sizes, int n_in,
                              void* d_out, int out_size, void* d_ws, size_t ws_size,
                              hipStream_t stream)
{
  const float* x   = (const float*)d_in[0];
  const float* pt  = (const float*)d_in[1];
  const float* w3d = (const float*)d_in[2];
  const float* b3d = (const float*)d_in[3];
  const float* w1  = (const float*)d_in[4];
  const float* b1  = (const float*)d_in[5];
  const float* g1  = (const float*)d_in[6];
  const float* be1 = (const float*)d_in[7];
  const float* w2  = (const float*)d_in[8];
  const float* b2  = (const float*)d_in[9];
  const float* g2  = (const float*)d_in[10];
  const float* be2 = (const float*)d_in[11];
  const float* w3  = (const float*)d_in[12];
  const float* b3  = (const float*)d_in[13];
  const float* g3  = (const float*)d_in[14];
  const float* be3 = (const float*)d_in[15];
  const float* wo  = (const float*)d_in[16];
  const float* bo  = (const float*)d_in[17];
  float* out = (float*)d_out;
  (void)in_sizes; (void)n_in; (void)out_size; (void)ws_size;

  char* ws = (char*)d_ws;
  size_t off = 0;
  auto take = [&](size_t nfl) {
    void* pp = ws + off;
    off = (off + nfl * sizeof(float) + 255) & ~(size_t)255;
    return (float*)pp;
  };
  float* feats  = take((size_t)BATCH * FEAT_ROWS * PVAL);   // ~37.7 MB
  float* glob   = take(BATCH * 32);
  float* mean   = take(256);
  float* invstd = take(256);
  float* yA     = take((size_t)BATCH * 256 * PVAL);         // y1 then y3 (~134 MB)
  float* yB     = take((size_t)BATCH * 128 * PVAL);         // y2 (~67 MB)
  float* wp1    = take(128 * 72);
  float* wp2    = take(128 * 128);
  float* wp3    = take(256 * 128);
  float* wpo    = take(32 * 256);

  pack_w_kernel<<< 36, 256, 0, stream>>>(w1, wp1, 128,  67, 128,  72);
  pack_w_kernel<<< 64, 256, 0, stream>>>(w2, wp2, 128, 128, 128, 128);
  pack_w_kernel<<<128, 256, 0, stream>>>(w3, wp3, 256, 128, 256, 128);
  pack_w_kernel<<< 32, 256, 0, stream>>>(wo, wpo,  10, 256,  32, 256);

  conv_gather_kernel<<<BATCH * PVAL / 256, 256, 0, stream>>>(x, pt, w3d, b3d, feats);
  glob_max_kernel<<<dim3(32, BATCH), 256, 0, stream>>>(feats, glob);
  glob_bcast_kernel<<<BATCH * 37 * PVAL / 256, 256, 0, stream>>>(glob, feats);

  // L1: 67(->72) -> 128   (MW=8: block 128M x 64N, X read once)
  gemm_lds_wmma<8><<<dim3(PVAL / 64, 1, BATCH), 256, 0, stream>>>(wp1, b1, feats, yA, 128, 72, FEAT_ROWS, 128);
  stats_kernel<<<128, 256, 0, stream>>>(yA, 128, mean, invstd);
  bnrelu_kernel<<<BATCH * 128 * PVAL / 256, 256, 0, stream>>>(yA, g1, be1, mean, invstd, 128, 128);
  // L2: 128 -> 128
  gemm_lds_wmma<8><<<dim3(PVAL / 64, 1, BATCH), 256, 0, stream>>>(wp2, b2, yA, yB, 128, 128, 128, 128);
  stats_kernel<<<128, 256, 0, stream>>>(yB, 128, mean, invstd);
  bnrelu_kernel<<<BATCH * 128 * PVAL / 256, 256, 0, stream>>>(yB, g2, be2, mean, invstd, 128, 128);
  // L3: 128 -> 256
  gemm_lds_wmma<8><<<dim3(PVAL / 64, 2, BATCH), 256, 0, stream>>>(wp3, b3, yB, yA, 256, 128, 128, 256);
  stats_kernel<<<256, 256, 0, stream>>>(yA, 256, mean, invstd);
  bnrelu_kernel<<<BATCH * 256 * PVAL / 256, 256, 0, stream>>>(yA, g3, be3, mean, invstd, 256, 256);
  // Output: 256 -> 10  (MW=2: block 32M x 256N, only 3.2x pad overhead)
  gemm_lds_wmma<2><<<dim3(PVAL / 256, 1, BATCH), 256, 0, stream>>>(wpo, bo, yA, out, 10, 256, 256, 10);
}